// RandDCGRUCell_15109694947762
// MI455X (gfx1250) — compile-verified
//
#include <hip/hip_runtime.h>
#include <math.h>

// ---------------------------------------------------------------------------
// RandDCGRU cell forward for MI455X (gfx1250), wave32 + V_WMMA_F32_16X16X4_F32
// ---------------------------------------------------------------------------
// B=64, N=2000, IN_DIM=2, U=64, K=2, D=66, M=5 diffusion matrices.
// Dominant cost: 8x dense S[2000x2000] @ X[2000x4224] fp32 matmuls (~270 GFLOP,
// working set ~200MB -> L2 resident; issue-limited, so maximize WMMA per load:
// 32x64 output tile per wave = 8 wmma per (2 b64 + 8 b32) loads per k-step.

typedef __attribute__((ext_vector_type(2))) float v2f;
typedef __attribute__((ext_vector_type(8))) float v8f;

#define NB   64           // batch
#define NN   2000         // nodes
#define UD   64           // hidden units
#define DDIM 66           // concat feature dim
#define MMAT 5            // diffusion matrices
#define LDX  (DDIM * NB)  // 4224 columns of diffusion operand
#define KP   336          // D*M = 330 padded up to multiple of 16
#define ROWS (NB * NN)    // 128000 rows of xcat

#define WMMA_F32(A, Bf, C) \
  __builtin_amdgcn_wmma_f32_16x16x4_f32(false, (A), false, (Bf), (short)0, (C), false, false)

// ------------------------------ elementwise --------------------------------

// x0[n*4224 + d*64 + b] = concat(inputs, state)[b, n, d]
__global__ void build_x0_kernel(const float* __restrict__ inp,
                                const float* __restrict__ st,
                                float* __restrict__ x0) {
  size_t i = (size_t)blockIdx.x * blockDim.x + threadIdx.x;
  const size_t total = (size_t)NN * LDX;
  if (i >= total) return;
  int b = (int)(i % NB);
  int d = (int)((i / NB) % DDIM);
  int n = (int)(i / LDX);
  float v;
  if (d < 2) v = inp[(size_t)b * (NN * 2) + (size_t)n * 2 + d];
  else       v = st[(size_t)b * (NN * UD) + (size_t)n * UD + (d - 2)];
  x0[i] = v;
}

// xcat[rid*336 + d*5+m] = xs[m][n*4224 + d*64 + b], rid = b*2000+n; pad K->336
__global__ void build_xcat_kernel(const float* __restrict__ xs,  // XS0 base, stride NN*LDX
                                  float* __restrict__ xcat) {
  size_t i = (size_t)blockIdx.x * blockDim.x + threadIdx.x;
  const size_t total = (size_t)ROWS * KP;
  if (i >= total) return;
  int kk  = (int)(i % KP);
  int rid = (int)(i / KP);
  float v = 0.0f;
  if (kk < DDIM * MMAT) {
    int d = kk / MMAT, m = kk % MMAT;
    int n = rid % NN,  b = rid / NN;
    v = xs[(size_t)m * ((size_t)NN * LDX) + (size_t)n * LDX + d * NB + b];
  }
  xcat[i] = v;
}

// reparameterized weights: w = mu + exp(log_sig)*eps, zero-padded rows 330..335
__global__ void prep_weights_kernel(const float* __restrict__ mwru, const float* __restrict__ lwru,
                                    const float* __restrict__ ewru,
                                    const float* __restrict__ mbru, const float* __restrict__ lbru,
                                    const float* __restrict__ ebru,
                                    const float* __restrict__ mwc,  const float* __restrict__ lwc,
                                    const float* __restrict__ ewc,
                                    const float* __restrict__ mbc,  const float* __restrict__ lbc,
                                    const float* __restrict__ ebc,
                                    float* __restrict__ wru, float* __restrict__ wc,
                                    float* __restrict__ bru, float* __restrict__ bc) {
  int i = blockIdx.x * blockDim.x + threadIdx.x;
  const int WRU_N = KP * 128, WC_N = KP * 64;
  if (i < WRU_N) {
    int k = i / 128, j = i % 128;
    wru[i] = (k < 330) ? mwru[k * 128 + j] + __expf(lwru[k * 128 + j]) * ewru[k * 128 + j] : 0.0f;
  } else if (i < WRU_N + WC_N) {
    int t = i - WRU_N; int k = t / 64, j = t % 64;
    wc[t] = (k < 330) ? mwc[k * 64 + j] + __expf(lwc[k * 64 + j]) * ewc[k * 64 + j] : 0.0f;
  } else if (i < WRU_N + WC_N + 128) {
    int j = i - (WRU_N + WC_N);
    bru[j] = mbru[j] + __expf(lbru[j]) * ebru[j];
  } else if (i < WRU_N + WC_N + 192) {
    int j = i - (WRU_N + WC_N + 128);
    bc[j] = mbc[j] + __expf(lbc[j]) * ebc[j];
  }
}

// rhx = sigmoid_r * hx  (value row-major [128000,128], r = cols 0..63)
__global__ void rhx_kernel(const float* __restrict__ value,
                           const float* __restrict__ hx,
                           float* __restrict__ rhx) {
  size_t i = (size_t)blockIdx.x * blockDim.x + threadIdx.x;
  if (i >= (size_t)ROWS * UD) return;
  size_t rid = i / UD; int j = (int)(i % UD);
  rhx[i] = value[rid * 128 + j] * hx[i];
}

// out = u*hx + (1-u)*c  (u = cols 64..127 of value)
__global__ void final_kernel(const float* __restrict__ value,
                             const float* __restrict__ hx,
                             const float* __restrict__ c,
                             float* __restrict__ out) {
  size_t i = (size_t)blockIdx.x * blockDim.x + threadIdx.x;
  if (i >= (size_t)ROWS * UD) return;
  size_t rid = i / UD; int j = (int)(i % UD);
  float u = value[rid * 128 + 64 + j];
  out[i] = u * hx[i] + (1.0f - u) * c[i];
}

// ------------------------- WMMA diffusion matmul ---------------------------
// Y = S @ X            (cheb == 0)
// Y = 2*(S @ X) - Z    (cheb == 1)
// S: [2000 x 2000], X/Y/Z: [2000 x 4224] row-major.
// Block = 64 threads (2 waves, split over columns). Wave tile = 32 rows x 64
// cols: 2 A-fragments + 4 B-fragments -> 8 WMMAs per k-step, B reused across
// both row fragments. Grid = (ceil(2000/32)=63, 4224/128=33). Row tail handled
// by clamping A-row reads (EXEC stays all-ones for WMMA) and predicating only
// the stores / Chebyshev Z reads.
__global__ __launch_bounds__(64) void diffuse_kernel(const float* __restrict__ S,
                                                     const float* __restrict__ X,
                                                     const float* __restrict__ Z,
                                                     float* __restrict__ Y,
                                                     int cheb) {
  const int lane = threadIdx.x & 31;
  const int wave = threadIdx.x >> 5;
  const int half = lane >> 4;   // A: K pair select; C/D: row-half select
  const int l15  = lane & 15;
  const int rowTile = blockIdx.x * 32;
  const int colBase = (blockIdx.y * 2 + wave) * 64;

  int ar0 = rowTile + l15;       if (ar0 > NN - 1) ar0 = NN - 1;
  int ar1 = rowTile + 16 + l15;  if (ar1 > NN - 1) ar1 = NN - 1;
  const float* Srow0 = S + (size_t)ar0 * NN;
  const float* Srow1 = S + (size_t)ar1 * NN;

  v8f c00 = {}, c01 = {}, c02 = {}, c03 = {};
  v8f c10 = {}, c11 = {}, c12 = {}, c13 = {};

  for (int k = 0; k < NN; k += 4) {
    const int ka = k + 2 * half;
    // A 16x4 fragments (two row tiles): lane {l15, half} holds S[row, ka..ka+1]
    v2f a0 = *(const v2f*)(Srow0 + ka);
    v2f a1 = *(const v2f*)(Srow1 + ka);
    // B 4x16 fragments: row ka across lanes 0..15 (.x), row ka+1 (.y)
    const float* x0p = X + (size_t)ka * LDX + colBase + l15;
    const float* x1p = x0p + LDX;
    v2f b0 = {x0p[0],  x1p[0]};
    v2f b1 = {x0p[16], x1p[16]};
    v2f b2 = {x0p[32], x1p[32]};
    v2f b3 = {x0p[48], x1p[48]};
    c00 = WMMA_F32(a0, b0, c00);
    c01 = WMMA_F32(a0, b1, c01);
    c02 = WMMA_F32(a0, b2, c02);
    c03 = WMMA_F32(a0, b3, c03);
    c10 = WMMA_F32(a1, b0, c10);
    c11 = WMMA_F32(a1, b1, c11);
    c12 = WMMA_F32(a1, b2, c12);
    c13 = WMMA_F32(a1, b3, c13);
  }

  // C/D layout: VGPR v, lane l -> (M = v + 8*half, N = l15)
  const int r0 = rowTile + 8 * half;
#pragma unroll
  for (int v = 0; v < 8; ++v) {
    int row0 = r0 + v;
    if (row0 < NN) {
      size_t off = (size_t)row0 * LDX + colBase + l15;
      float y0 = c00[v], y1 = c01[v], y2 = c02[v], y3 = c03[v];
      if (cheb) {
        y0 = 2.0f * y0 - Z[off +  0];
        y1 = 2.0f * y1 - Z[off + 16];
        y2 = 2.0f * y2 - Z[off + 32];
        y3 = 2.0f * y3 - Z[off + 48];
      }
      Y[off +  0] = y0; Y[off + 16] = y1; Y[off + 32] = y2; Y[off + 48] = y3;
    }
    int row1 = r0 + 16 + v;
    if (row1 < NN) {
      size_t off = (size_t)row1 * LDX + colBase + l15;
      float y0 = c10[v], y1 = c11[v], y2 = c12[v], y3 = c13[v];
      if (cheb) {
        y0 = 2.0f * y0 - Z[off +  0];
        y1 = 2.0f * y1 - Z[off + 16];
        y2 = 2.0f * y2 - Z[off + 32];
        y3 = 2.0f * y3 - Z[off + 48];
      }
      Y[off +  0] = y0; Y[off + 16] = y1; Y[off + 32] = y2; Y[off + 48] = y3;
    }
  }
}

// ---------------------- WMMA projection GEMM + act -------------------------
// Out[rid, j] = act( xcat[rid, :] @ W[:, j] + bias[j] )
// xcat: [128000 x 336] (K zero-padded), W: [336 x outdim] (rows padded zero).
// Wave tile = 32 rows x 64 cols; 128000/32 = 4000 row tiles exactly.
// grid = (4000, 1), block = 32*(outdim/64) threads (waves split columns).
__global__ __launch_bounds__(64) void proj_kernel(const float* __restrict__ A,
                                                  const float* __restrict__ W,
                                                  const float* __restrict__ bias,
                                                  float* __restrict__ Out,
                                                  int outdim, int act /*0=sigmoid,1=tanh*/) {
  const int lane = threadIdx.x & 31;
  const int wave = threadIdx.x >> 5;
  const int half = lane >> 4;
  const int l15  = lane & 15;
  const int rowTile = blockIdx.x * 32;
  const int colBase = wave * 64;

  const float* Arow0 = A + (size_t)(rowTile + l15) * KP;
  const float* Arow1 = A + (size_t)(rowTile + 16 + l15) * KP;

  v8f c00 = {}, c01 = {}, c02 = {}, c03 = {};
  v8f c10 = {}, c11 = {}, c12 = {}, c13 = {};

  for (int k = 0; k < KP; k += 4) {
    const int ka = k + 2 * half;
    v2f a0 = *(const v2f*)(Arow0 + ka);
    v2f a1 = *(const v2f*)(Arow1 + ka);
    const float* w0 = W + (size_t)ka * outdim + colBase + l15;
    const float* w1 = w0 + outdim;
    v2f b0 = {w0[0],  w1[0]};
    v2f b1 = {w0[16], w1[16]};
    v2f b2 = {w0[32], w1[32]};
    v2f b3 = {w0[48], w1[48]};
    c00 = WMMA_F32(a0, b0, c00);
    c01 = WMMA_F32(a0, b1, c01);
    c02 = WMMA_F32(a0, b2, c02);
    c03 = WMMA_F32(a0, b3, c03);
    c10 = WMMA_F32(a1, b0, c10);
    c11 = WMMA_F32(a1, b1, c11);
    c12 = WMMA_F32(a1, b2, c12);
    c13 = WMMA_F32(a1, b3, c13);
  }

  const float bb0 = bias[colBase + l15];
  const float bb1 = bias[colBase + l15 + 16];
  const float bb2 = bias[colBase + l15 + 32];
  const float bb3 = bias[colBase + l15 + 48];
  const int r0 = rowTile + 8 * half;
#pragma unroll
  for (int v = 0; v < 8; ++v) {
#pragma unroll
    for (int t = 0; t < 2; ++t) {
      int row = r0 + 16 * t + v;
      size_t off = (size_t)row * outdim + colBase + l15;
      float y0 = (t ? c10[v] : c00[v]) + bb0;
      float y1 = (t ? c11[v] : c01[v]) + bb1;
      float y2 = (t ? c12[v] : c02[v]) + bb2;
      float y3 = (t ? c13[v] : c03[v]) + bb3;
      if (act == 0) {
        y0 = 1.0f / (1.0f + __expf(-y0));
        y1 = 1.0f / (1.0f + __expf(-y1));
        y2 = 1.0f / (1.0f + __expf(-y2));
        y3 = 1.0f / (1.0f + __expf(-y3));
      } else {
        y0 = tanhf(y0); y1 = tanhf(y1); y2 = tanhf(y2); y3 = tanhf(y3);
      }
      Out[off +  0] = y0; Out[off + 16] = y1; Out[off + 32] = y2; Out[off + 48] = y3;
    }
  }
}

// --------------------------------- driver ----------------------------------

extern "C" void kernel_launch(void* const* d_in, const int* in_sizes, int n_in,
                              void* d_out, int out_size, void* d_ws, size_t ws_size,
                              hipStream_t stream) {
  const float* inputs   = (const float*)d_in[0];
  const float* hx       = (const float*)d_in[1];
  const float* supports = (const float*)d_in[2];
  const float* mwru = (const float*)d_in[3];
  const float* lwru = (const float*)d_in[4];
  const float* mbru = (const float*)d_in[5];
  const float* lbru = (const float*)d_in[6];
  const float* ewru = (const float*)d_in[7];
  const float* ebru = (const float*)d_in[8];
  const float* mwc  = (const float*)d_in[9];
  const float* lwc  = (const float*)d_in[10];
  const float* mbc  = (const float*)d_in[11];
  const float* lbc  = (const float*)d_in[12];
  const float* ewc  = (const float*)d_in[13];
  const float* ebc  = (const float*)d_in[14];
  float* out = (float*)d_out;

  // workspace layout (floats)
  float* ws = (float*)d_ws;
  const size_t XSsz = (size_t)NN * LDX;             // 8,448,000
  float* XS0  = ws;
  float* XS1  = XS0 + XSsz;
  float* XS2  = XS1 + XSsz;
  float* XS3  = XS2 + XSsz;
  float* XS4  = XS3 + XSsz;
  float* XCAT = XS4 + XSsz;                         // 128000 * 336
  float* VAL  = XCAT + (size_t)ROWS * KP;           // 128000 * 128
  float* CBUF = VAL  + (size_t)ROWS * 128;          // 128000 * 64
  float* RHX  = CBUF + (size_t)ROWS * 64;           // 128000 * 64
  float* WRU  = RHX  + (size_t)ROWS * 64;           // 336 * 128
  float* WC   = WRU  + (size_t)KP * 128;            // 336 * 64
  float* BRU  = WC   + (size_t)KP * 64;             // 128
  float* BC   = BRU  + 128;                         // 64

  const float* S0 = supports;
  const float* S1 = supports + (size_t)NN * NN;

  const dim3 diffGrid((NN + 31) / 32, LDX / 128);   // (63, 33)
  const dim3 diffBlk(64);
  const int  x0Blocks   = (int)(((size_t)NN * LDX + 255) / 256);
  const int  xcatBlocks = (int)(((size_t)ROWS * KP + 255) / 256);
  const int  ewBlocks   = (int)(((size_t)ROWS * UD + 255) / 256);

  prep_weights_kernel<<<(KP * 192 + 192 + 255) / 256, 256, 0, stream>>>(
      mwru, lwru, ewru, mbru, lbru, ebru, mwc, lwc, ewc, mbc, lbc, ebc,
      WRU, WC, BRU, BC);

  // -------- gconv 1: gates r,u --------
  build_x0_kernel<<<x0Blocks, 256, 0, stream>>>(inputs, hx, XS0);
  diffuse_kernel<<<diffGrid, diffBlk, 0, stream>>>(S0, XS0, nullptr, XS1, 0);
  diffuse_kernel<<<diffGrid, diffBlk, 0, stream>>>(S0, XS1, XS0, XS2, 1);
  diffuse_kernel<<<diffGrid, diffBlk, 0, stream>>>(S1, XS0, nullptr, XS3, 0);
  diffuse_kernel<<<diffGrid, diffBlk, 0, stream>>>(S1, XS3, XS0, XS4, 1);
  build_xcat_kernel<<<xcatBlocks, 256, 0, stream>>>(XS0, XCAT);
  proj_kernel<<<dim3(ROWS / 32, 1), 64, 0, stream>>>(XCAT, WRU, BRU, VAL, 128, 0);
  rhx_kernel<<<ewBlocks, 256, 0, stream>>>(VAL, hx, RHX);

  // -------- gconv 2: candidate c --------
  build_x0_kernel<<<x0Blocks, 256, 0, stream>>>(inputs, RHX, XS0);
  diffuse_kernel<<<diffGrid, diffBlk, 0, stream>>>(S0, XS0, nullptr, XS1, 0);
  diffuse_kernel<<<diffGrid, diffBlk, 0, stream>>>(S0, XS1, XS0, XS2, 1);
  diffuse_kernel<<<diffGrid, diffBlk, 0, stream>>>(S1, XS0, nullptr, XS3, 0);
  diffuse_kernel<<<diffGrid, diffBlk, 0, stream>>>(S1, XS3, XS0, XS4, 1);
  build_xcat_kernel<<<xcatBlocks, 256, 0, stream>>>(XS0, XCAT);
  proj_kernel<<<dim3(ROWS / 32, 1), 32, 0, stream>>>(XCAT, WC, BC, CBUF, 64, 1);

  // -------- GRU blend --------
  final_kernel<<<ewBlocks, 256, 0, stream>>>(VAL, hx, CBUF, out);
}